// GCBlock_76570676953218
// MI455X (gfx1250) — compile-verified
//
#include <hip/hip_runtime.h>

#define F 64
#define PSTR 160  // pair-row stride (dwords) for packed W in LDS: conflict-free b64 B loads
#define WSZ (32 * PSTR)  // 5120 dwords per packed 64x64 weight matrix
#define STG 68   // staging row stride (dwords): conflict-free stores + A-layout readback
#define STGSZ (16 * STG) // 1088 dwords per-wave staging tile

typedef __attribute__((ext_vector_type(2))) float v2f;
typedef __attribute__((ext_vector_type(8))) float v8f;

// Same-wave LDS fence: DS ops are in-order per wave; this blocks compiler
// reordering and waits for outstanding DS ops so staged data is visible.
__device__ __forceinline__ void lds_fence_wave() {
  asm volatile("s_wait_dscnt 0" ::: "memory");
}

// Pack W[64][64] (row-major, K-major) into pair-interleaved LDS layout:
// Wp[(k>>1)*PSTR + n*2 + (k&1)] = W[k][n]  -> B fragment = one aligned b64.
__device__ __forceinline__ void pack_w(float* dst, const float* __restrict__ src,
                                       int tid, int nthr) {
  for (int idx = tid; idx < F * F; idx += nthr) {
    const int k = idx >> 6, n = idx & 63;
    dst[(k >> 1) * PSTR + n * 2 + (k & 1)] = src[idx];
  }
}

// acc[0..3] (16x16 f32 C-layout tiles covering N=0..63) = A[16x64] * W[64x64]
// A fragments: lane m16 holds row m16; hi=0 lanes hold K pair {4t,4t+1},
// hi=1 lanes hold {4t+2,4t+3}. B fragments mirror that layout (packed LDS).
__device__ __forceinline__ void wave_gemm(const v2f (&a)[16], const float* Wp,
                                          int m16, int hi, v8f (&acc)[4]) {
  v8f zero = {};
#pragma unroll
  for (int nt = 0; nt < 4; ++nt) acc[nt] = zero;
#pragma unroll
  for (int t = 0; t < 16; ++t) {
    const int kp = 2 * t + hi;  // packed pair-row = (4t+2hi)>>1
#pragma unroll
    for (int nt = 0; nt < 4; ++nt) {
      const v2f b =
          *reinterpret_cast<const v2f*>(&Wp[kp * PSTR + (nt * 16 + m16) * 2]);
      acc[nt] = __builtin_amdgcn_wmma_f32_16x16x4_f32(
          false, a[t], false, b, (short)0, acc[nt], false, false);
    }
  }
}

// C-layout -> row-major (stride STG) 16x64 tile in per-wave LDS staging.
__device__ __forceinline__ void stage_acc(const v8f (&acc)[4], float* stg,
                                          int m16, int hi) {
#pragma unroll
  for (int nt = 0; nt < 4; ++nt)
#pragma unroll
    for (int v = 0; v < 8; ++v)
      stg[(v + 8 * hi) * STG + nt * 16 + m16] = acc[nt][v];
}

// Coalesced 16x64 tile store: LDS staging (stride STG) -> global (row-major).
__device__ __forceinline__ void store_tile(const float* stg, float* dst,
                                           int lane) {
#pragma unroll
  for (int i = 0; i < 8; ++i) {
    const int j = i * 128 + lane * 4;           // 0..1023, float4 granules
    const int l = (j >> 6) * STG + (j & 63);    // padded LDS index (16B aligned)
    *reinterpret_cast<float4*>(dst + j) =
        *reinterpret_cast<const float4*>(stg + l);
  }
}

// ---------------- pxh = px @ W_pp (rows = N * X, row-major) ----------------
__global__ void pp_kernel(const float* __restrict__ px,
                          const float* __restrict__ W,
                          float* __restrict__ out, int nRows) {
  extern __shared__ float lds[];
  pack_w(lds, W, threadIdx.x, blockDim.x);
  __syncthreads();

  const int lane = threadIdx.x & 31;
  const int wave = threadIdx.x >> 5;
  const int m16 = lane & 15, hi = lane >> 4;
  const int nTiles = nRows >> 4;
  const int tile = blockIdx.x * (blockDim.x >> 5) + wave;
  if (tile >= nTiles) return;  // wave-uniform guard; EXEC stays all-1s

  const long long r0 = (long long)tile * 16;
  const float* src = px + (r0 + m16) * F;
  v2f a[16];
#pragma unroll
  for (int t = 0; t < 16; ++t)
    a[t] = *reinterpret_cast<const v2f*>(src + 4 * t + 2 * hi);

  v8f acc[4];
  wave_gemm(a, lds, m16, hi, acc);
  float* dst = out + r0 * F;
#pragma unroll
  for (int nt = 0; nt < 4; ++nt)
#pragma unroll
    for (int v = 0; v < 8; ++v)
      dst[(v + 8 * hi) * F + nt * 16 + m16] = acc[nt][v];
}

// ---------------- rank-1 invariant branch (edges) ----------------
__global__ void edge1_kernel(const int* __restrict__ pair_i,
                             const int* __restrict__ pair_j,
                             const float* __restrict__ basis,
                             const float* __restrict__ p1h,
                             const float* __restrict__ Wpi0,
                             const float* __restrict__ Wpi1,
                             const float* __restrict__ Wii,
                             float* __restrict__ out_i1,
                             float* __restrict__ out_p1, int E_) {
  extern __shared__ float lds[];
  float* W0 = lds;
  float* W1 = lds + WSZ;
  float* W2 = lds + 2 * WSZ;
  float* stgBase = lds + 3 * WSZ;                   // 4 waves * STGSZ
  int* idxBase = (int*)(lds + 3 * WSZ + 4 * STGSZ); // 4 waves * 16 ints
  pack_w(W0, Wpi0, threadIdx.x, blockDim.x);
  pack_w(W1, Wpi1, threadIdx.x, blockDim.x);
  pack_w(W2, Wii, threadIdx.x, blockDim.x);
  __syncthreads();

  const int lane = threadIdx.x & 31;
  const int wave = threadIdx.x >> 5;
  const int m16 = lane & 15, hi = lane >> 4;
  float* stg = stgBase + wave * STGSZ;
  int* idxBuf = idxBase + wave * 16;
  const int nTiles = E_ >> 4;
  const int tile = blockIdx.x * (blockDim.x >> 5) + wave;
  if (tile >= nTiles) return;

  const long long e0 = (long long)tile * 16;
  const long long eRow = e0 + m16;
  const long long ii = pair_i[eRow];
  const long long jj = pair_j[eRow];
  if (hi == 0) idxBuf[m16] = (int)ii;  // scatter target rows, one per edge-row
  const float* si = p1h + ii * F;
  const float* sj = p1h + jj * F;
  const float* sb = basis + eRow * F;

  v2f a[16];
#pragma unroll
  for (int t = 0; t < 16; ++t) {
    const int k = 4 * t + 2 * hi;
    const v2f x = *reinterpret_cast<const v2f*>(si + k);
    const v2f y = *reinterpret_cast<const v2f*>(sj + k);
    const v2f z = *reinterpret_cast<const v2f*>(sb + k);
    a[t] = x + y + z;
  }

  v8f acc[4];
  // inter @ W_pi1_0
  wave_gemm(a, W0, m16, hi, acc);
  stage_acc(acc, stg, m16, hi);
  lds_fence_wave();
  // tanh, reload in A-layout
#pragma unroll
  for (int t = 0; t < 16; ++t) {
    const v2f x =
        *reinterpret_cast<const v2f*>(&stg[m16 * STG + 4 * t + 2 * hi]);
    a[t].x = tanhf(x.x);
    a[t].y = tanhf(x.y);
  }
  // @ W_pi1_1 -> i1
  wave_gemm(a, W1, m16, hi, acc);
  stage_acc(acc, stg, m16, hi);
  lds_fence_wave();
  store_tile(stg, out_i1 + e0 * F, lane);  // coalesced i1 store
#pragma unroll
  for (int t = 0; t < 16; ++t)
    a[t] = *reinterpret_cast<const v2f*>(&stg[m16 * STG + 4 * t + 2 * hi]);
  // ii = i1 @ W_ii1, scatter-add to p1_out
  wave_gemm(a, W2, m16, hi, acc);
#pragma unroll
  for (int v = 0; v < 8; ++v) {
    const long long dstRow = (long long)idxBuf[v + 8 * hi] * F;
#pragma unroll
    for (int nt = 0; nt < 4; ++nt)
      atomicAdd(&out_p1[dstRow + nt * 16 + m16], acc[nt][v]);
  }
}

// ---------------- rank-X equivariant branch (edges), X = 3 or 5 ----------------
template <int X>
__global__ void edgeX_kernel(const int* __restrict__ pair_i,
                             const int* __restrict__ pair_j,
                             const float* __restrict__ basis,
                             const float* __restrict__ pxh,
                             const float* __restrict__ diff,
                             const float* __restrict__ Wpi,
                             const float* __restrict__ Wii,
                             float* __restrict__ out_ix,
                             float* __restrict__ out_px, int E_) {
  extern __shared__ float lds[];
  float* W0 = lds;
  float* W1 = lds + WSZ;
  float* stgBase = lds + 2 * WSZ;
  int* idxBase = (int*)(lds + 2 * WSZ + 4 * STGSZ);
  pack_w(W0, Wpi, threadIdx.x, blockDim.x);
  pack_w(W1, Wii, threadIdx.x, blockDim.x);
  __syncthreads();

  const int lane = threadIdx.x & 31;
  const int wave = threadIdx.x >> 5;
  const int m16 = lane & 15, hi = lane >> 4;
  float* stg = stgBase + wave * STGSZ;
  int* idxBuf = idxBase + wave * 16;
  const int nRows = E_ * X;
  const int nTiles = nRows >> 4;
  const int tile = blockIdx.x * (blockDim.x >> 5) + wave;
  if (tile >= nTiles) return;

  const long long r0 = (long long)tile * 16;
  const long long row = r0 + m16;  // row = e*X + x
  const long long e = row / X;
  const int x = (int)(row - e * X);
  const long long ii = pair_i[e];
  const long long jj = pair_j[e];
  if (hi == 0) idxBuf[m16] = (int)(ii * X + x);  // scatter target row
  const float* si = pxh + (ii * X + x) * F;
  const float* sj = pxh + (jj * X + x) * F;
  const float* sb = basis + e * F;

  v2f a[16];
#pragma unroll
  for (int t = 0; t < 16; ++t) {
    const int k = 4 * t + 2 * hi;
    const v2f xx = *reinterpret_cast<const v2f*>(si + k);
    const v2f yy = *reinterpret_cast<const v2f*>(sj + k);
    const v2f zz = *reinterpret_cast<const v2f*>(sb + k);
    a[t] = xx + yy + zz;
  }

  v8f acc[4];
  // (gather+basis) @ W_pi
  wave_gemm(a, W0, m16, hi, acc);
  stage_acc(acc, stg, m16, hi);
  lds_fence_wave();
  // scale by diff[e, x] (one scalar per row = per lane), reload A-layout
  const float dr = diff[row];  // diff is [E, X, 1] flat = row index
#pragma unroll
  for (int t = 0; t < 16; ++t) {
    const v2f xx =
        *reinterpret_cast<const v2f*>(&stg[m16 * STG + 4 * t + 2 * hi]);
    a[t].x = xx.x * dr;
    a[t].y = xx.y * dr;
  }
  // @ W_ii -> ix
  wave_gemm(a, W1, m16, hi, acc);
  // scatter-add to px_out from C-layout (targets via LDS broadcast)
#pragma unroll
  for (int v = 0; v < 8; ++v) {
    const long long dstRow = (long long)idxBuf[v + 8 * hi] * F;
#pragma unroll
    for (int nt = 0; nt < 4; ++nt)
      atomicAdd(&out_px[dstRow + nt * 16 + m16], acc[nt][v]);
  }
  // coalesced ix store via staging
  stage_acc(acc, stg, m16, hi);
  lds_fence_wave();
  store_tile(stg, out_ix + r0 * F, lane);
}

extern "C" void kernel_launch(void* const* d_in, const int* in_sizes, int n_in,
                              void* d_out, int out_size, void* d_ws, size_t ws_size,
                              hipStream_t stream) {
  const int* pair_i = (const int*)d_in[0];
  const int* pair_j = (const int*)d_in[1];
  const float* basis = (const float*)d_in[2];
  const float* p1 = (const float*)d_in[3];
  const float* p3 = (const float*)d_in[4];
  const float* p5 = (const float*)d_in[5];
  const float* norm_diff = (const float*)d_in[6];
  const float* diff_p5 = (const float*)d_in[7];
  const float* W_pp1 = (const float*)d_in[8];
  const float* W_pi1_0 = (const float*)d_in[9];
  const float* W_pi1_1 = (const float*)d_in[10];
  const float* W_ii1 = (const float*)d_in[11];
  const float* W_pp3 = (const float*)d_in[12];
  const float* W_pi3 = (const float*)d_in[13];
  const float* W_ii3 = (const float*)d_in[14];
  const float* W_pp5 = (const float*)d_in[15];
  const float* W_pi5 = (const float*)d_in[16];
  const float* W_ii5 = (const float*)d_in[17];

  const int E_ = in_sizes[0];
  const int N_ = in_sizes[3] / F;

  // workspace: pxh tables (fit in L2: 115 MB total)
  float* p1h = (float*)d_ws;
  float* p3h = p1h + (size_t)N_ * F;
  float* p5h = p3h + (size_t)N_ * 3 * F;

  // output layout (concatenated, return order)
  float* p1_out = (float*)d_out;
  float* i1 = p1_out + (size_t)N_ * F;
  float* p3_out = i1 + (size_t)E_ * F;
  float* i3 = p3_out + (size_t)N_ * 3 * F;
  float* p5_out = i3 + (size_t)E_ * 3 * F;
  float* i5 = p5_out + (size_t)N_ * 5 * F;

  // zero the scatter-add targets
  hipMemsetAsync(p1_out, 0, (size_t)N_ * F * sizeof(float), stream);
  hipMemsetAsync(p3_out, 0, (size_t)N_ * 3 * F * sizeof(float), stream);
  hipMemsetAsync(p5_out, 0, (size_t)N_ * 5 * F * sizeof(float), stream);

  const int BLOCK = 128;  // 4 wave32 waves
  auto blocksFor = [](int rows) {
    int tiles = (rows + 15) / 16;
    return (tiles + 3) / 4;
  };

  const size_t ppLds = (size_t)WSZ * sizeof(float);  // 20 KB
  pp_kernel<<<blocksFor(N_), BLOCK, ppLds, stream>>>(p1, W_pp1, p1h, N_);
  pp_kernel<<<blocksFor(3 * N_), BLOCK, ppLds, stream>>>(p3, W_pp3, p3h, 3 * N_);
  pp_kernel<<<blocksFor(5 * N_), BLOCK, ppLds, stream>>>(p5, W_pp5, p5h, 5 * N_);

  const size_t e1Lds = (3 * (size_t)WSZ + 4 * STGSZ + 64) * sizeof(float);  // ~77 KB
  edge1_kernel<<<blocksFor(E_), BLOCK, e1Lds, stream>>>(
      pair_i, pair_j, basis, p1h, W_pi1_0, W_pi1_1, W_ii1, i1, p1_out, E_);

  const size_t exLds = (2 * (size_t)WSZ + 4 * STGSZ + 64) * sizeof(float);  // ~57 KB
  edgeX_kernel<3><<<blocksFor(3 * E_), BLOCK, exLds, stream>>>(
      pair_i, pair_j, basis, p3h, norm_diff, W_pi3, W_ii3, i3, p3_out, E_);
  edgeX_kernel<5><<<blocksFor(5 * E_), BLOCK, exLds, stream>>>(
      pair_i, pair_j, basis, p5h, diff_p5, W_pi5, W_ii5, i5, p5_out, E_);
}